// EF_el_22067541967355
// MI455X (gfx1250) — compile-verified
//
#include <hip/hip_runtime.h>

// Problem constants (from reference): B=8, T=16, CH=1, H=W=256
#define NB 8
#define NT 16
#define NH 256
#define NW 256
#define HW (NH * NW)

// GCC-style vector types matching the async-LDS builtins' expected pointees.
typedef int v2i_t __attribute__((vector_size(2 * sizeof(int))));
typedef int v4i_t __attribute__((vector_size(4 * sizeof(int))));

// ---------------------------------------------------------------------------
// Workspace layout (bytes):
//   [0,     8MB)  float4 state[NB*HW] : {X, Y, R, pad} per point
//   [8MB,  12MB)  float2 accw[NB*HW]  : {acc, wacc} splat buffer (L2-resident)
// ---------------------------------------------------------------------------

__global__ void pc_init_kernel(const float* __restrict__ x_in,
                               float4* __restrict__ state,
                               float2* __restrict__ accw) {
    const int p = blockIdx.x * blockDim.x + threadIdx.x;   // 0..HW-1
    const int b = blockIdx.y;                              // 0..NB-1
    const int i = p / NW;
    const int j = p - i * NW;
    float4 s;
    s.x = (float)j * (1.0f / (NW - 1));                    // meshgrid X
    s.y = (float)i * (1.0f / (NH - 1));                    // meshgrid Y
    s.z = x_in[((size_t)b * NT + (NT - 1)) * HW + p];      // R_pc0 = x[b,T-1]
    s.w = 0.0f;
    state[(size_t)b * HW + p] = s;
    accw[(size_t)b * HW + p]  = make_float2(0.0f, 0.0f);
}

__global__ void pc_step_kernel(const float* __restrict__ dec,
                               float4* __restrict__ state,
                               float2* __restrict__ accw,
                               int t) {
    const int p = blockIdx.x * blockDim.x + threadIdx.x;
    const int b = blockIdx.y;

    float4 s = state[(size_t)b * HW + p];                  // global_load_b128
    float X = s.x, Y = s.y, R = s.z;

    // ---- bilinear gather of (U,V,C) at (X*W-0.5, Y*H-0.5), zero padded ----
    // Branch-free: clamped indices + predicated weights (v_cndmask, no jumps).
    const float ix = X * (float)NW - 0.5f;                 // in [-0.5, 255.5]
    const float iy = Y * (float)NH - 0.5f;
    const float x0 = floorf(ix), y0 = floorf(iy);
    const float fx = ix - x0,    fy = iy - y0;
    const int x0i = (int)x0, y0i = (int)y0;                // in [-1, 255]

    const float wx0 = (x0i >= 0)       ? (1.0f - fx) : 0.0f;
    const float wx1 = (x0i + 1 <= NW-1) ? fx          : 0.0f;
    const float wy0 = (y0i >= 0)       ? (1.0f - fy) : 0.0f;
    const float wy1 = (y0i + 1 <= NH-1) ? fy          : 0.0f;
    const int xc0 = max(x0i, 0),     xc1 = min(x0i + 1, NW - 1);
    const int yc0 = max(y0i, 0) * NW, yc1 = min(y0i + 1, NH - 1) * NW;

    const float* __restrict__ plane = dec + (size_t)((b * NT + t) * 3) * HW;
    const int i00 = yc0 + xc0, i01 = yc0 + xc1, i10 = yc1 + xc0, i11 = yc1 + xc1;
    const float w00 = wy0 * wx0, w01 = wy0 * wx1, w10 = wy1 * wx0, w11 = wy1 * wx1;

    float u = w00 * plane[i00] + w01 * plane[i01] + w10 * plane[i10] + w11 * plane[i11];
    float v = w00 * plane[HW + i00] + w01 * plane[HW + i01]
            + w10 * plane[HW + i10] + w11 * plane[HW + i11];
    float c = w00 * plane[2*HW + i00] + w01 * plane[2*HW + i01]
            + w10 * plane[2*HW + i10] + w11 * plane[2*HW + i11];
    const float wsum = w00 + w01 + w10 + w11;
    // Affine transform (d-0.5)*5/{W,H} applied under zero-padded weighting:
    // sum(w*(d-0.5))*k = (sum(w*d) - 0.5*sum(w))*k
    u = (u - 0.5f * wsum) * (5.0f / (float)NW);
    v = (v - 0.5f * wsum) * (5.0f / (float)NH);

    X = fminf(fmaxf(X + u, 0.0f), 1.0f);
    Y = fminf(fmaxf(Y + v, 0.0f), 1.0f);
    R = fminf(fmaxf(R * c, 0.0f), 1.0f);
    s.x = X; s.y = Y; s.z = R;
    state[(size_t)b * HW + p] = s;                         // global_store_b128

    // Warm L2 with next timestep's planes (gfx1250 global_prefetch_b8).
    // Points drift <= ~2.5px/step, so same-index prefetch is a good predictor.
    if (t + 1 < NT) {
        const float* np = dec + (size_t)(((b * NT) + t + 1) * 3) * HW;
        __builtin_prefetch(np + p);
        __builtin_prefetch(np + HW + p);
        __builtin_prefetch(np + 2 * HW + p);
    }

    // ---- bilinear splat of R into {acc, wacc} ----
    // After the clip, sx,sy in [0, 255]; a tap can only be OOB when
    // sx0 == 255, which forces gx == 0 (zero weight). So clamped-index
    // unconditional atomics are exactly equivalent to the reference's
    // w*valid masking: OOB taps atomically add 0.0f.
    const float sx  = X * (float)(NW - 1);
    const float sy  = Y * (float)(NH - 1);
    const float sx0 = floorf(sx), sy0 = floorf(sy);
    const float gx  = sx - sx0,   gy  = sy - sy0;
    const int jx0 = (int)sx0,            jy0 = ((int)sy0) * NW;
    const int jx1 = min((int)sx0 + 1, NW - 1);
    const int jy1 = min(((int)sy0 + 1), NH - 1) * NW;
    const float swx0 = 1.0f - gx, swy0 = 1.0f - gy;

    float* __restrict__ aw = reinterpret_cast<float*>(accw + (size_t)b * HW);
    const float v00 = swy0 * swx0, v01 = swy0 * gx, v10 = gy * swx0, v11 = gy * gx;
    atomicAdd(aw + 2*(jy0 + jx0),     R * v00);
    atomicAdd(aw + 2*(jy0 + jx0) + 1, v00);
    atomicAdd(aw + 2*(jy0 + jx1),     R * v01);
    atomicAdd(aw + 2*(jy0 + jx1) + 1, v01);
    atomicAdd(aw + 2*(jy1 + jx0),     R * v10);
    atomicAdd(aw + 2*(jy1 + jx0) + 1, v10);
    atomicAdd(aw + 2*(jy1 + jx1),     R * v11);
    atomicAdd(aw + 2*(jy1 + jx1) + 1, v11);
}

__global__ void pc_norm_kernel(float2* __restrict__ accw,
                               float* __restrict__ out,
                               int t) {
    __shared__ float4 tile[256];                 // {acc0,w0,acc1,w1} per lane
    const int tid = threadIdx.x;
    const int idx = blockIdx.x * 512 + tid * 2;  // 2 points per thread
    const int b   = blockIdx.y;
    float4* src = reinterpret_cast<float4*>(accw + (size_t)b * HW + idx);

#if __has_builtin(__builtin_amdgcn_global_load_async_to_lds_b128)
    // gfx1250 async data mover: memory -> LDS (b128), tracked by ASYNCcnt.
    __builtin_amdgcn_global_load_async_to_lds_b128(
        (__attribute__((address_space(1))) v4i_t*)src,
        (__attribute__((address_space(3))) v4i_t*)&tile[tid],
        0, 0);
#if __has_builtin(__builtin_amdgcn_s_wait_asynccnt)
    __builtin_amdgcn_s_wait_asynccnt(0);
#else
    asm volatile("s_wait_asynccnt 0" ::: "memory");
#endif
    const float4 a = tile[tid];
#else
    const float4 a = *src;                       // fallback: plain b128 load
#endif

    float2 r;
    r.x = a.x / fmaxf(a.y, 1e-8f);
    r.y = a.z / fmaxf(a.w, 1e-8f);
    *reinterpret_cast<float2*>(out + ((size_t)b * NT + t) * HW + idx) = r;
    *src = make_float4(0.0f, 0.0f, 0.0f, 0.0f);  // reset for next step
}

extern "C" void kernel_launch(void* const* d_in, const int* in_sizes, int n_in,
                              void* d_out, int out_size, void* d_ws, size_t ws_size,
                              hipStream_t stream) {
    (void)in_sizes; (void)n_in; (void)out_size; (void)ws_size;
    const float* x   = (const float*)d_in[0];   // (B,T,1,H,W) f32
    const float* dec = (const float*)d_in[1];   // (B,T,3,H,W) f32
    float* out = (float*)d_out;                 // (B,T,1,H,W) f32

    float4* state = (float4*)d_ws;
    float2* accw  = (float2*)((char*)d_ws + (size_t)NB * HW * sizeof(float4));

    const dim3 block(256, 1, 1);
    const dim3 gridA(HW / 256, NB, 1);
    const dim3 gridN(HW / 512, NB, 1);

    pc_init_kernel<<<gridA, block, 0, stream>>>(x, state, accw);
    for (int t = 0; t < NT; ++t) {
        pc_step_kernel<<<gridA, block, 0, stream>>>(dec, state, accw, t);
        pc_norm_kernel<<<gridN, block, 0, stream>>>(accw, out, t);
    }
}